// SyntheticTreePropagationNetwork_14353780703997
// MI455X (gfx1250) — compile-verified
//
#include <hip/hip_runtime.h>
#include <hip/hip_bf16.h>

// Problem constants (match reference)
#define B_TREES 2048
#define N_NODES 32
#define D_DIM   256
#define V_OUT   1026            // V + 2
#define ROWS    (B_TREES * N_NODES)   // 65536
#define EPS_BN  1e-5f

typedef __attribute__((ext_vector_type(2))) float v2f;
typedef __attribute__((ext_vector_type(8))) float v8f;

// ---------------------------------------------------------------------------
// fp32 WMMA 16x16x4 fragment mapping (per CDNA5 ISA):
//   A frag (16x4 f32, 2 VGPR): lane l -> m = l&15, k-pair = (l>>4)*2 {+0,+1}
//   B frag (4x16 f32, 2 VGPR): lane l -> n = l&15, k-pair = (l>>4)*2 {+0,+1}
//   C/D   (16x16 f32, 8 VGPR): vgpr r -> m = (l>>4)*8 + r, n = l&15
// A is staged in LDS K-major: As[k*16 + m] -> conflict-free ds reads.
// ---------------------------------------------------------------------------
__device__ __forceinline__ v8f wmma_f32_tile(const float* __restrict__ As,
                                             const float* __restrict__ Bmat,
                                             int ldb, int n0)
{
    const int lane = threadIdx.x & 31;
    const int m    = lane & 15;
    const int kk   = (lane >> 4) * 2;
    v8f c = {};
    #pragma unroll 4
    for (int k0 = 0; k0 < D_DIM; k0 += 4) {
        v2f a, b;
        a.x = As[(k0 + kk) * 16 + m];
        a.y = As[(k0 + kk + 1) * 16 + m];
        b.x = Bmat[(size_t)(k0 + kk) * ldb + n0 + m];
        b.y = Bmat[(size_t)(k0 + kk + 1) * ldb + n0 + m];
        c = __builtin_amdgcn_wmma_f32_16x16x4_f32(false, a, false, b,
                                                  (short)0, c, false, false);
    }
    return c;
}

// ---------------------------------------------------------------------------
// K0: zero BN accumulators (ws is poisoned 0xAA by the harness).
// ---------------------------------------------------------------------------
__global__ void zero_stats_kernel(float* stats) {
    stats[blockIdx.x * blockDim.x + threadIdx.x] = 0.0f;
}

// ---------------------------------------------------------------------------
// K1: sibling sums (+bp folded in) and zero z root row.
// grid = B_TREES blocks of D_DIM threads (thread = feature d).
// ---------------------------------------------------------------------------
__global__ void sibling_kernel(const float* __restrict__ mol,
                               const int* __restrict__ parent,
                               const float* __restrict__ bp,
                               float* __restrict__ sib,
                               float* __restrict__ z)
{
    __shared__ float mols[N_NODES * D_DIM];   // 32 KB
    __shared__ int   par[N_NODES];
    const int b = blockIdx.x;
    const int d = threadIdx.x;
    if (d < N_NODES) par[d] = parent[b * N_NODES + d];
    for (int j = 0; j < N_NODES; ++j)
        mols[j * D_DIM + d] = mol[((size_t)b * N_NODES + j) * D_DIM + d];
    __syncthreads();

    z[(size_t)b * N_NODES * D_DIM + d] = 0.0f;   // root row z[b][0][:] = 0
    const float bpv = bp[d];
    for (int t = 1; t < N_NODES; ++t) {
        const int pt = par[t];
        float acc = bpv;
        for (int j = 0; j < t; ++j)
            if (par[j] == pt) acc += mols[j * D_DIM + d];
        sib[((size_t)b * N_NODES + t) * D_DIM + d] = acc;
    }
}

// ---------------------------------------------------------------------------
// K2: tree-propagation scan. Block = 512 threads (16 waves) owns 16 trees.
// z kept in global (streams through 192MB L2); agent-scope ld/st so the
// same-WGP cross-step RAW bypasses the read-only WGP$.
// Release across the step boundary = s_wait_storecnt 0 + workgroup barrier:
// SCOPE_DEV stores only decrement STOREcnt once they reach L2 (the agent
// coherence point), and the next step's gathers are SCOPE_DEV loads that
// read L2 directly. This avoids __threadfence()'s global_wb + global_inv,
// which would evict the hot Wp/sib lines from the WGP$ every step.
// ---------------------------------------------------------------------------
__global__ void __launch_bounds__(512)
scan_kernel(const float* __restrict__ sib,
            const int* __restrict__ parent,
            const float* __restrict__ Wp,
            float* __restrict__ z)
{
    __shared__ float As[D_DIM * 16];        // A^T staging, 16 KB
    __shared__ int   par_s[16 * N_NODES];
    const int b0 = blockIdx.x * 16;
    for (int i = threadIdx.x; i < 16 * N_NODES; i += blockDim.x)
        par_s[i] = parent[(size_t)(b0 + (i >> 5)) * N_NODES + (i & 31)];
    __syncthreads();

    const int wave = threadIdx.x >> 5;
    const int lane = threadIdx.x & 31;
    const int n    = lane & 15;
    const int half = lane >> 4;
    const int n0   = wave * 16;

    for (int t = 1; t < N_NODES; ++t) {
        // stage A = relu(z[tree][parent_t][:]) -> As[k*16 + m]
        for (int i = threadIdx.x; i < 16 * D_DIM; i += blockDim.x) {
            const int m = i >> 8;          // tree in block
            const int k = i & (D_DIM - 1);
            const int p = par_s[m * N_NODES + t];
            float v = __hip_atomic_load(
                &z[((size_t)(b0 + m) * N_NODES + p) * D_DIM + k],
                __ATOMIC_RELAXED, __HIP_MEMORY_SCOPE_AGENT);
            As[k * 16 + m] = fmaxf(v, 0.0f);
        }
        __syncthreads();

        // batched sib prefetch (independent of the GEMM below)
        float sv[8];
        #pragma unroll
        for (int r = 0; r < 8; ++r) {
            const int m = half * 8 + r;
            sv[r] = sib[((size_t)(b0 + m) * N_NODES + t) * D_DIM + n0 + n];
        }

        v8f c = wmma_f32_tile(As, Wp, D_DIM, n0);

        #pragma unroll
        for (int r = 0; r < 8; ++r) {
            const int m = half * 8 + r;
            const size_t idx = ((size_t)(b0 + m) * N_NODES + t) * D_DIM + n0 + n;
            __hip_atomic_store(&z[idx], c[r] + sv[r], __ATOMIC_RELAXED,
                               __HIP_MEMORY_SCOPE_AGENT);
        }
        // Lightweight agent-scope release: stores have reached L2 when
        // STOREcnt hits zero (SCOPE_DEV semantics). No cache wb/inv needed.
        asm volatile("s_wait_storecnt 0x0" ::: "memory");
        __syncthreads();
    }
}

// ---------------------------------------------------------------------------
// K3/K6: per-feature sum & sumsq partials (256 rows per block).
// ---------------------------------------------------------------------------
#define BN_ROWS_PER_BLOCK 256
__global__ void bn_stats_kernel(const float* __restrict__ x,
                                float* __restrict__ sum,
                                float* __restrict__ sumsq)
{
    const int d  = threadIdx.x;                      // 256 features
    const int r0 = blockIdx.x * BN_ROWS_PER_BLOCK;
    float s = 0.0f, q = 0.0f;
    for (int r = 0; r < BN_ROWS_PER_BLOCK; ++r) {
        const float v = x[(size_t)(r0 + r) * D_DIM + d];
        s += v; q += v * v;
    }
    atomicAdd(&sum[d], s);
    atomicAdd(&sumsq[d], q);
}

// Fold BN into per-feature scale a and shift c.
__global__ void bn_finalize_kernel(const float* __restrict__ sum,
                                   const float* __restrict__ sumsq,
                                   const float* __restrict__ gamma,
                                   const float* __restrict__ beta,
                                   float* __restrict__ a,
                                   float* __restrict__ c)
{
    const int d = threadIdx.x;
    const float inv_n = 1.0f / (float)ROWS;
    const float mu  = sum[d] * inv_n;
    const float var = sumsq[d] * inv_n - mu * mu;
    const float s   = gamma[d] * rsqrtf(var + EPS_BN);
    a[d] = s;
    c[d] = beta[d] - mu * s;
}

// ---------------------------------------------------------------------------
// K5/K8: Y[r,:] = relu(X[r,:]*a + c) @ W + bias, X is [ROWS x 256].
// Block = 512 threads (16 waves) owns 16*MT rows. Register blocking MT x NT:
// each A fragment (LDS) is reused across NT column tiles and each B fragment
// (global) across MT row tiles -> inner k-step is 2*MT ds + 2*NT vmem loads
// per MT*NT WMMAs. Tail columns use a clamped, memory-safe index and are
// never stored (a WMMA output column depends only on the same B column).
// ---------------------------------------------------------------------------
template<int MT, int NT>
__global__ void __launch_bounds__(512)
bn_relu_gemm_kernel(const float* __restrict__ X,
                    const float* __restrict__ a,
                    const float* __restrict__ cshift,
                    const float* __restrict__ W,     // [256 x Ncols] row-major
                    const float* __restrict__ bias,  // [Ncols]
                    float* __restrict__ Y,           // [ROWS x Ncols]
                    int Ncols)
{
    __shared__ float As[MT * D_DIM * 16];            // MT * 16 KB
    const int r0 = blockIdx.x * (16 * MT);
    for (int i = threadIdx.x; i < 16 * MT * D_DIM; i += blockDim.x) {
        const int m = i >> 8;
        const int k = i & (D_DIM - 1);
        const float v = X[(size_t)(r0 + m) * D_DIM + k];
        As[(m >> 4) * (D_DIM * 16) + k * 16 + (m & 15)] =
            fmaxf(fmaf(v, a[k], cshift[k]), 0.0f);
    }
    __syncthreads();

    const int wave = threadIdx.x >> 5;
    const int lane = threadIdx.x & 31;
    const int n    = lane & 15;
    const int half = lane >> 4;
    const int kk   = half * 2;
    const int ntiles  = (Ncols + 15) >> 4;
    const int ngroups = (ntiles + NT - 1) / NT;

    for (int grp = wave; grp < ngroups; grp += 16) { // wave-uniform bound
        bool  nok[NT];
        int   ncol[NT];
        float bb[NT];
        #pragma unroll
        for (int nt = 0; nt < NT; ++nt) {
            const int tile = grp * NT + nt;
            const int col  = tile * 16 + n;
            nok[nt]  = (tile < ntiles) && (col < Ncols);
            ncol[nt] = nok[nt] ? col : (Ncols - 1);  // clamped, safe address
            bb[nt]   = bias[ncol[nt]];
        }

        v8f acc[MT][NT];
        #pragma unroll
        for (int mt = 0; mt < MT; ++mt)
            #pragma unroll
            for (int nt = 0; nt < NT; ++nt) acc[mt][nt] = {};

        #pragma unroll 2
        for (int k0 = 0; k0 < D_DIM; k0 += 4) {
            v2f b[NT];
            #pragma unroll
            for (int nt = 0; nt < NT; ++nt) {
                b[nt].x = W[(size_t)(k0 + kk) * Ncols + ncol[nt]];
                b[nt].y = W[(size_t)(k0 + kk + 1) * Ncols + ncol[nt]];
            }
            #pragma unroll
            for (int mt = 0; mt < MT; ++mt) {
                v2f av;
                av.x = As[mt * (D_DIM * 16) + (k0 + kk) * 16 + n];
                av.y = As[mt * (D_DIM * 16) + (k0 + kk + 1) * 16 + n];
                #pragma unroll
                for (int nt = 0; nt < NT; ++nt)
                    acc[mt][nt] = __builtin_amdgcn_wmma_f32_16x16x4_f32(
                        false, av, false, b[nt], (short)0, acc[mt][nt],
                        false, false);
            }
        }

        #pragma unroll
        for (int nt = 0; nt < NT; ++nt) {
            if (nok[nt]) {
                const int tile = grp * NT + nt;
                #pragma unroll
                for (int mt = 0; mt < MT; ++mt) {
                    #pragma unroll
                    for (int r = 0; r < 8; ++r) {
                        const int m = mt * 16 + half * 8 + r;
                        Y[(size_t)(r0 + m) * Ncols + tile * 16 + n] =
                            acc[mt][nt][r] + bb[nt];
                    }
                }
            }
        }
    }
}

#define GEMM_MT 4                                      // 64 rows per block

// ---------------------------------------------------------------------------
// Launch. Inputs: 0 mol_vec, 1 parent_idx, 2 Wp, 3 bp, 4 bn1_g, 5 bn1_b,
//                 6 W1, 7 b1, 8 bn2_g, 9 bn2_b, 10 W2, 11 b2.
// Workspace: z (64MB) | sib reused as h (64MB) | stats (4KB) | coefs (4KB)
// ---------------------------------------------------------------------------
extern "C" void kernel_launch(void* const* d_in, const int* in_sizes, int n_in,
                              void* d_out, int out_size, void* d_ws, size_t ws_size,
                              hipStream_t stream)
{
    const float* mol    = (const float*)d_in[0];
    const int*   parent = (const int*)  d_in[1];
    const float* Wp     = (const float*)d_in[2];
    const float* bp     = (const float*)d_in[3];
    const float* bn1_g  = (const float*)d_in[4];
    const float* bn1_b  = (const float*)d_in[5];
    const float* W1     = (const float*)d_in[6];
    const float* b1     = (const float*)d_in[7];
    const float* bn2_g  = (const float*)d_in[8];
    const float* bn2_b  = (const float*)d_in[9];
    const float* W2     = (const float*)d_in[10];
    const float* b2     = (const float*)d_in[11];
    float* out = (float*)d_out;

    const size_t ZN = (size_t)ROWS * D_DIM;          // 16,777,216 floats
    float* zbuf  = (float*)d_ws;
    float* sib_h = zbuf + ZN;                        // sib, later reused as h
    float* sum1  = sib_h + ZN;
    float* sq1   = sum1 + D_DIM;
    float* sum2  = sq1  + D_DIM;
    float* sq2   = sum2 + D_DIM;
    float* a1    = sq2  + D_DIM;
    float* c1    = a1   + D_DIM;
    float* a2    = c1   + D_DIM;
    float* c2    = a2   + D_DIM;

    // K0: zero the four BN accumulators (4 * 256 floats)
    zero_stats_kernel<<<4, D_DIM, 0, stream>>>(sum1);

    // K1: sibling sums (+bp), zero z root rows
    sibling_kernel<<<B_TREES, D_DIM, 0, stream>>>(mol, parent, bp, sib_h, zbuf);

    // K2: sequential tree scan, 16 trees per block
    scan_kernel<<<B_TREES / 16, 512, 0, stream>>>(sib_h, parent, Wp, zbuf);

    // K3/K4: BN1 stats + fold
    bn_stats_kernel<<<ROWS / BN_ROWS_PER_BLOCK, D_DIM, 0, stream>>>(zbuf, sum1, sq1);
    bn_finalize_kernel<<<1, D_DIM, 0, stream>>>(sum1, sq1, bn1_g, bn1_b, a1, c1);

    // K5: h = relu(bn1(z)) @ W1 + b1   (h overwrites sib)
    //     16 column tiles -> NT=1 keeps all 16 waves busy.
    bn_relu_gemm_kernel<GEMM_MT, 1><<<ROWS / (16 * GEMM_MT), 512, 0, stream>>>(
        zbuf, a1, c1, W1, b1, sib_h, D_DIM);

    // K6/K7: BN2 stats + fold
    bn_stats_kernel<<<ROWS / BN_ROWS_PER_BLOCK, D_DIM, 0, stream>>>(sib_h, sum2, sq2);
    bn_finalize_kernel<<<1, D_DIM, 0, stream>>>(sum2, sq2, bn2_g, bn2_b, a2, c2);

    // K8: logits = relu(bn2(h)) @ W2 + b2
    //     65 column tiles -> NT=2 (33 tile pairs over 16 waves), halves LDS A
    //     fragment traffic and lifts WMMAs per load to MT*NT per k-step.
    bn_relu_gemm_kernel<GEMM_MT, 2><<<ROWS / (16 * GEMM_MT), 512, 0, stream>>>(
        sib_h, a2, c2, W2, b2, out, V_OUT);
}